// TTAttention_36447092474252
// MI455X (gfx1250) — compile-verified
//
#include <hip/hip_runtime.h>
#include <math.h>

// TT-attention for MI455X (gfx1250, wave32, WMMA + Tensor Data Mover).
// All matrix math uses V_WMMA_F32_16X16X4_F32 (exact fp32 path).
// All LDS tile staging uses TENSOR_LOAD_TO_LDS (TDM), double-buffered,
// with D# pad_interval/pad_amount reproducing the bank-conflict padding.
//
// Workspace layout (floats):
//   [0 .. 4M)    : W_q, W_k, W_v, W_o   (4 x 1024x1024)
//   [4M .. 8M)   : Q projected          (4096 x 1024)
//   [8M .. 12M)  : K projected
//   [12M .. 16M) : V projected
//   [16M .. 20M) : attention context
// Total 20M floats = 80 MB.

typedef __attribute__((ext_vector_type(2))) float v2f;
typedef __attribute__((ext_vector_type(8))) float v8f;
typedef __attribute__((ext_vector_type(4))) unsigned int v4u;
typedef __attribute__((ext_vector_type(8))) int v8i;
typedef __attribute__((ext_vector_type(4))) int v4i;

#define DM 1024
#define SEQ 2048
#define NB 2
#define NH 16
#define DH 64

// ---------------------------------------------------------------------------
// TDM helper: 2D tile load (global -> LDS), fp32 elements, row-padded LDS.
//   tile_x elements per row (contiguous), tile_y rows, row stride in elements.
//   pad_icode: pad after (2 << pad_icode) DWORDs; pad_acode: pad (acode+1) DWORDs.
// D# layout per CDNA5 ISA sec 8.3-8.5 (groups 2/3 zero => 2D tensor).
// This toolchain exposes the 6-arg builtin: (v4u, v8i, v4i, v4i, v8i, cpol).
// ---------------------------------------------------------------------------
__device__ __forceinline__ unsigned lds_off(const void* p) {
  // Generic LDS pointer: low 32 bits = byte offset within the WG's LDS segment.
  return (unsigned)(uintptr_t)p;
}

__device__ __forceinline__ void tdm_load_2d(unsigned lds_addr, const float* gptr,
                                            unsigned tile_x, unsigned tile_y,
                                            unsigned stride_el,
                                            unsigned pad_icode, unsigned pad_acode) {
  unsigned long long ga = (unsigned long long)(uintptr_t)gptr;
  v4u g0;
  g0[0] = 1u;                                         // count=1, gather off
  g0[1] = lds_addr;                                   // lds_addr (bytes)
  g0[2] = (unsigned)(ga & 0xFFFFFFFFu);               // global_addr[31:0]
  g0[3] = (unsigned)((ga >> 32) & 0x01FFFFFFu)        // global_addr[56:32]
          | (2u << 30);                               // type = 2 ("image")
  v8i g1;
  g1[0] = (int)((2u << 16)                            // data_size = 4 bytes
                | (1u << 20)                          // pad_enable
                | (pad_icode << 22)                   // pad_interval
                | (pad_acode << 25));                 // pad_amount
  g1[1] = (int)(tile_x << 16);                        // tensor_dim0[15:0]
  g1[2] = (int)(tile_y << 16);                        // dim0 hi=0 | tensor_dim1[15:0]
  g1[3] = (int)(tile_x << 16);                        // dim1 hi=0 | tile_dim0
  g1[4] = (int)tile_y;                                // tile_dim1 | tile_dim2=0
  g1[5] = (int)stride_el;                             // tensor_dim0_stride[31:0]
  g1[6] = 0;                                          // stride0 hi | stride1 lo
  g1[7] = 0;
  v4i gz4 = {0, 0, 0, 0};
  v8i gz8 = {0, 0, 0, 0, 0, 0, 0, 0};
  __builtin_amdgcn_tensor_load_to_lds(g0, g1, gz4, gz4, gz8, 0);
}

// ---------------------------------------------------------------------------
// Kernel 1: dense weight from TT cores.
// W[o, i] = sum_r g0[0, i1, j1, r] * g1[r, i2, j2, 0]
// ---------------------------------------------------------------------------
__global__ __launch_bounds__(256) void tt_build_w(const float* __restrict__ g0,
                                                  const float* __restrict__ g1,
                                                  float* __restrict__ W) {
  int idx = blockIdx.x * 256 + threadIdx.x;  // 0 .. 1024*1024-1
  int o  = idx >> 10;
  int in = idx & 1023;
  int i1 = o >> 5, i2 = o & 31;
  int j1 = in >> 5, j2 = in & 31;
  const float* a = g0 + (i1 * 32 + j1) * 8;
  float s = 0.f;
#pragma unroll
  for (int r = 0; r < 8; ++r)
    s += a[r] * g1[(r * 32 + i2) * 32 + j2];
  W[idx] = s;
}

// ---------------------------------------------------------------------------
// Kernel 2: Y[4096,1024] = X[4096,1024] @ W[1024,1024]^T + bias
// 8 waves; 128(M) x 64(N) WG tile; K-tile 32 TDM-staged, double-buffered.
// ---------------------------------------------------------------------------
#define LDA 36  // 32 data + 4 pad DWORDs; 36r mod 64 distinct over 16 rows

__global__ __launch_bounds__(256) void tt_gemm_bias(const float* __restrict__ X,
                                                    const float* __restrict__ W,
                                                    const float* __restrict__ bias,
                                                    float* __restrict__ Y) {
  __shared__ __attribute__((aligned(16))) float As[2][128 * LDA];
  __shared__ __attribute__((aligned(16))) float Bs[2][64 * LDA];

  const int tid  = threadIdx.x;
  const int wave = tid >> 5;
  const int lane = tid & 31;
  const int half = lane >> 4;
  const int l15  = lane & 15;
  const int m0   = blockIdx.y * 128;
  const int n0   = blockIdx.x * 64;

  const float* Abase = X + (size_t)m0 * DM;
  const float* Bbase = W + (size_t)n0 * DM;

  v8f acc[4] = {};

  if (wave == 0) {
    tdm_load_2d(lds_off(&As[0][0]), Abase, 32, 128, DM, 4, 3);
    tdm_load_2d(lds_off(&Bs[0][0]), Bbase, 32, 64, DM, 4, 3);
  }

  for (int it = 0; it < DM / 32; ++it) {
    const int cur = it & 1;
    if (wave == 0) __builtin_amdgcn_s_wait_tensorcnt((short)0);
    __syncthreads();  // tiles[cur] visible to all waves
    if (wave == 0 && it + 1 < DM / 32) {
      const int k0n = (it + 1) * 32;
      tdm_load_2d(lds_off(&As[cur ^ 1][0]), Abase + k0n, 32, 128, DM, 4, 3);
      tdm_load_2d(lds_off(&Bs[cur ^ 1][0]), Bbase + k0n, 32, 64, DM, 4, 3);
    }

    const float* ap = &As[cur][(wave * 16 + l15) * LDA + 2 * half];
#pragma unroll
    for (int kk = 0; kk < 32; kk += 4) {
      v2f a = *(const v2f*)(ap + kk);  // A: row = 16w + l15, K = kk + 2*half .. +1
#pragma unroll
      for (int t = 0; t < 4; ++t) {
        // B = W^T : b.x = W[n0+col][kk+2*half]
        v2f b = *(const v2f*)&Bs[cur][(t * 16 + l15) * LDA + kk + 2 * half];
        acc[t] = __builtin_amdgcn_wmma_f32_16x16x4_f32(
            false, a, false, b, (short)0, acc[t], false, false);
      }
    }
    __syncthreads();  // readers done (waits DScnt) before buffer reuse
  }

#pragma unroll
  for (int t = 0; t < 4; ++t) {
    int col = n0 + t * 16 + l15;
    float bv = bias[col];
#pragma unroll
    for (int v = 0; v < 8; ++v) {
      int row = m0 + wave * 16 + v + 8 * half;  // C/D layout: M = v + 8*half
      Y[(size_t)row * DM + col] = acc[t][v] + bv;
    }
  }
}

// ---------------------------------------------------------------------------
// Kernel 3: flash attention per (b,h). 8 waves, 128 Q rows per WG.
// KV tiles of 32 TDM-staged (double-buffered), online softmax, fp32 WMMA.
// ---------------------------------------------------------------------------
#define LDK 68  // 64 data + 4 pad DWORDs; 4r mod 64 distinct over 16 rows
#define LDP 34  // even (float2 reads); scalar stores

__global__ __launch_bounds__(256) void tt_attn(const float* __restrict__ Qp,
                                               const float* __restrict__ Kp,
                                               const float* __restrict__ Vp,
                                               float* __restrict__ Ctx) {
  __shared__ __attribute__((aligned(16))) float Ks[2][32 * LDK];
  __shared__ __attribute__((aligned(16))) float Vs[2][32 * LDK];
  __shared__ __attribute__((aligned(16))) float Ps[8 * 16 * LDP];

  const int tid  = threadIdx.x;
  const int wave = tid >> 5;
  const int lane = tid & 31;
  const int half = lane >> 4;
  const int l15  = lane & 15;
  const int bh = blockIdx.y;
  const int b = bh >> 4, h = bh & 15;
  const int q0 = blockIdx.x * 128;

  const size_t batchoff = (size_t)b * SEQ * DM;
  const size_t headoff  = (size_t)h * DH;

  // Q tile (16 rows/wave) into registers in A-layout, pre-scaled by 1/sqrt(64).
  v2f aq[16];
  {
    const float* qrow =
        Qp + batchoff + (size_t)(q0 + wave * 16 + l15) * DM + headoff + 2 * half;
#pragma unroll
    for (int i = 0; i < 16; ++i) {
      v2f q = *(const v2f*)(qrow + 4 * i);
      aq[i] = q * 0.125f;
    }
  }

  v8f acc_o[4] = {};
  float m_i[8], l_i[8];
#pragma unroll
  for (int v = 0; v < 8; ++v) { m_i[v] = -1e30f; l_i[v] = 0.f; }

  if (wave == 0) {
    size_t g = batchoff + headoff;
    tdm_load_2d(lds_off(&Ks[0][0]), Kp + g, 64, 32, DM, 5, 3);
    tdm_load_2d(lds_off(&Vs[0][0]), Vp + g, 64, 32, DM, 5, 3);
  }

  for (int it = 0; it < SEQ / 32; ++it) {
    const int cur = it & 1;
    if (wave == 0) __builtin_amdgcn_s_wait_tensorcnt((short)0);
    __syncthreads();  // K/V tiles[cur] visible
    if (wave == 0 && it + 1 < SEQ / 32) {
      size_t g = batchoff + (size_t)((it + 1) * 32) * DM + headoff;
      tdm_load_2d(lds_off(&Ks[cur ^ 1][0]), Kp + g, 64, 32, DM, 5, 3);
      tdm_load_2d(lds_off(&Vs[cur ^ 1][0]), Vp + g, 64, 32, DM, 5, 3);
    }

    // S = (Q/8) K^T : 16 x 32 per wave
    v8f acc_s[2] = {};
#pragma unroll
    for (int i = 0; i < 16; ++i) {
      v2f a = aq[i];
#pragma unroll
      for (int t = 0; t < 2; ++t) {
        // B = K^T: b.x = K[j0 + t*16 + col][4i + 2*half]
        v2f bk = *(const v2f*)&Ks[cur][(t * 16 + l15) * LDK + 4 * i + 2 * half];
        acc_s[t] = __builtin_amdgcn_wmma_f32_16x16x4_f32(
            false, a, false, bk, (short)0, acc_s[t], false, false);
      }
    }

    // Online softmax: row r = v + 8*half spans the 16 lanes of this half-group.
#pragma unroll
    for (int v = 0; v < 8; ++v) {
      float rm = fmaxf(acc_s[0][v], acc_s[1][v]);
      rm = fmaxf(rm, __shfl_xor(rm, 1, 32));
      rm = fmaxf(rm, __shfl_xor(rm, 2, 32));
      rm = fmaxf(rm, __shfl_xor(rm, 4, 32));
      rm = fmaxf(rm, __shfl_xor(rm, 8, 32));
      float mnew  = fmaxf(m_i[v], rm);
      float alpha = __expf(m_i[v] - mnew);
      float p0 = __expf(acc_s[0][v] - mnew);
      float p1 = __expf(acc_s[1][v] - mnew);
      acc_s[0][v] = p0;
      acc_s[1][v] = p1;
      float rs = p0 + p1;
      rs += __shfl_xor(rs, 1, 32);
      rs += __shfl_xor(rs, 2, 32);
      rs += __shfl_xor(rs, 4, 32);
      rs += __shfl_xor(rs, 8, 32);
      l_i[v] = l_i[v] * alpha + rs;
      m_i[v] = mnew;
#pragma unroll
      for (int t = 0; t < 4; ++t) acc_o[t][v] *= alpha;
    }

    // P (D-layout) -> per-wave LDS slice -> reload in A-layout.
    float* pw = &Ps[wave * 16 * LDP];
#pragma unroll
    for (int t = 0; t < 2; ++t)
#pragma unroll
      for (int v = 0; v < 8; ++v)
        pw[(v + 8 * half) * LDP + t * 16 + l15] = acc_s[t][v];
    // Same-wave DS RAW: DS ops are in-order per wave; wait + compiler fence.
    asm volatile("s_wait_dscnt 0" ::: "memory");

    // O += P @ V
#pragma unroll
    for (int i = 0; i < 8; ++i) {
      v2f a = *(const v2f*)&pw[l15 * LDP + 4 * i + 2 * half];
#pragma unroll
      for (int t = 0; t < 4; ++t) {
        v2f bv;
        bv[0] = Vs[cur][(4 * i + 2 * half) * LDK + t * 16 + l15];
        bv[1] = Vs[cur][(4 * i + 2 * half + 1) * LDK + t * 16 + l15];
        acc_o[t] = __builtin_amdgcn_wmma_f32_16x16x4_f32(
            false, a, false, bv, (short)0, acc_o[t], false, false);
      }
    }
    __syncthreads();  // readers done before buffer reuse
  }

  // Normalize and write context.
#pragma unroll
  for (int t = 0; t < 4; ++t) {
#pragma unroll
    for (int v = 0; v < 8; ++v) {
      int row = q0 + wave * 16 + v + 8 * half;
      Ctx[batchoff + (size_t)row * DM + headoff + t * 16 + l15] =
          acc_o[t][v] / l_i[v];
    }
  }
}

// ---------------------------------------------------------------------------
// Launch
// ---------------------------------------------------------------------------
extern "C" void kernel_launch(void* const* d_in, const int* in_sizes, int n_in,
                              void* d_out, int out_size, void* d_ws, size_t ws_size,
                              hipStream_t stream) {
  (void)in_sizes; (void)n_in; (void)out_size; (void)ws_size;

  const float* query = (const float*)d_in[0];
  const float* key   = (const float*)d_in[1];
  const float* value = (const float*)d_in[2];
  const float* g0[4] = {(const float*)d_in[3], (const float*)d_in[6],
                        (const float*)d_in[9], (const float*)d_in[12]};
  const float* g1[4] = {(const float*)d_in[4], (const float*)d_in[7],
                        (const float*)d_in[10], (const float*)d_in[13]};
  const float* bs[4] = {(const float*)d_in[5], (const float*)d_in[8],
                        (const float*)d_in[11], (const float*)d_in[14]};

  float* ws = (float*)d_ws;
  const size_t MW = (size_t)DM * DM;        // 1M floats per weight
  const size_t MX = (size_t)NB * SEQ * DM;  // 4M floats per activation
  float* W[4];
  for (int i = 0; i < 4; ++i) W[i] = ws + i * MW;
  float* Qp  = ws + 4 * MW;
  float* Kp  = Qp + MX;
  float* Vp  = Kp + MX;
  float* Ctx = Vp + MX;

  // 1) TT -> dense weights
  for (int i = 0; i < 4; ++i)
    tt_build_w<<<(DM * DM) / 256, 256, 0, stream>>>(g0[i], g1[i], W[i]);

  // 2) Q/K/V projections
  dim3 ggrid(DM / 64, (NB * SEQ) / 128);  // (16, 32)
  tt_gemm_bias<<<ggrid, 256, 0, stream>>>(query, W[0], bs[0], Qp);
  tt_gemm_bias<<<ggrid, 256, 0, stream>>>(key,   W[1], bs[1], Kp);
  tt_gemm_bias<<<ggrid, 256, 0, stream>>>(value, W[2], bs[2], Vp);

  // 3) attention
  dim3 agrid(SEQ / 128, NB * NH);  // (16, 32)
  tt_attn<<<agrid, 256, 0, stream>>>(Qp, Kp, Vp, Ctx);

  // 4) output projection
  tt_gemm_bias<<<ggrid, 256, 0, stream>>>(Ctx, W[3], bs[3], (float*)d_out);
}